// EdgeConv_16037407884013
// MI455X (gfx1250) — compile-verified
//
#include <hip/hip_runtime.h>

typedef float v2f __attribute__((ext_vector_type(2)));
typedef float v8f __attribute__((ext_vector_type(8)));

#define DFEAT 128

// ---------------------------------------------------------------------------
// Node-level GEMMs on the WMMA units (full fp32 via V_WMMA_F32_16X16X4_F32):
//   T = feat @ theta_w
//   A = T + feat @ phi_w + theta_b + phi_b
// Block = 16 rows x 128 cols, 8 wave32s (one 16x16 tile per wave), K = 128.
// ---------------------------------------------------------------------------
__global__ void __launch_bounds__(256) edgeconv_node_gemm(
    const float* __restrict__ feat,
    const float* __restrict__ theta_w,
    const float* __restrict__ theta_b,
    const float* __restrict__ phi_w,
    const float* __restrict__ phi_b,
    float* __restrict__ T,
    float* __restrict__ A,
    int n_nodes)
{
    const int wave   = threadIdx.x >> 5;
    const int lane   = threadIdx.x & 31;
    const int lane16 = lane & 15;
    const int hi     = lane >> 4;          // half-wave selector (K offset 0 / 2)
    const int row0   = blockIdx.x * 16;
    const int n0     = wave * 16;          // this wave's output column tile

    // A-fragment row for this lane (16x4 fp32 A layout: lanes 0-15 K=0,1; 16-31 K=2,3)
    int aRow = row0 + lane16;
    if (aRow >= n_nodes) aRow = n_nodes - 1;   // clamp (N=50000 is 16-aligned; safety)
    const float* aPtr = feat + (size_t)aRow * DFEAT;

    const int col = n0 + lane16;

    v8f accT = {};
    v8f accP = {};

#pragma unroll 4
    for (int k0 = 0; k0 < DFEAT; k0 += 4) {
        const int ka = k0 + 2 * hi;
        v2f a = *(const v2f*)(aPtr + ka);

        v2f bt, bp;
        bt.x = theta_w[(size_t)ka * DFEAT + col];
        bt.y = theta_w[(size_t)(ka + 1) * DFEAT + col];
        bp.x = phi_w[(size_t)ka * DFEAT + col];
        bp.y = phi_w[(size_t)(ka + 1) * DFEAT + col];

        accT = __builtin_amdgcn_wmma_f32_16x16x4_f32(
            false, a, false, bt, (short)0, accT, false, false);
        accP = __builtin_amdgcn_wmma_f32_16x16x4_f32(
            false, a, false, bp, (short)0, accP, false, false);
    }

    const float bias = theta_b[col] + phi_b[col];

    // C/D layout: VGPR r -> row M=r (lanes 0-15) and M=r+8 (lanes 16-31)
    float* tBase = T + (size_t)(row0 + 8 * hi) * DFEAT + col;
    float* aBase = A + (size_t)(row0 + 8 * hi) * DFEAT + col;

    if (row0 + 16 <= n_nodes) {
        // uniform fast path: no per-row predication
#pragma unroll
        for (int r = 0; r < 8; ++r) {
            const float t = accT[r];
            tBase[(size_t)r * DFEAT] = t;
            aBase[(size_t)r * DFEAT] = t + accP[r] + bias;
        }
    } else {
#pragma unroll
        for (int r = 0; r < 8; ++r) {
            const int row = row0 + r + 8 * hi;
            if (row < n_nodes) {
                const float t = accT[r];
                T[(size_t)row * DFEAT + col] = t;
                A[(size_t)row * DFEAT + col] = t + accP[r] + bias;
            }
        }
    }
}

// ---------------------------------------------------------------------------
// Segment-min scatter.  out[n] = A[n] - min_{edges e: dst[e]==n} T[src[e]].
// Floats mapped to order-preserving u32 keys so atomicMin(u32) implements
// float min exactly (no NaNs in this workload).
// ---------------------------------------------------------------------------
__device__ __forceinline__ unsigned f32_to_key(float x) {
    unsigned u = __float_as_uint(x);
    return (u & 0x80000000u) ? ~u : (u | 0x80000000u);
}
__device__ __forceinline__ float key_to_f32(unsigned k) {
    unsigned u = (k & 0x80000000u) ? (k & 0x7FFFFFFFu) : ~k;
    return __uint_as_float(u);
}

__global__ void __launch_bounds__(256) edgeconv_init_keys(
    uint4* __restrict__ keys, int n4)
{
    const int i = blockIdx.x * 256 + threadIdx.x;
    if (i < n4)
        keys[i] = make_uint4(0xFFFFFFFFu, 0xFFFFFFFFu, 0xFFFFFFFFu, 0xFFFFFFFFu);
}

// 32 lanes per edge; each lane gathers a float4 of T[src] (global_load_b128,
// 512B coalesced per edge) and issues 4 u32 atomic-mins into keys[dst].
__global__ void __launch_bounds__(256) edgeconv_edge_min(
    const float* __restrict__ T,
    const int* __restrict__ src,
    const int* __restrict__ dst,
    unsigned* __restrict__ keys,
    int n_edges)
{
    const int idx = blockIdx.x * 256 + threadIdx.x;
    const int e = idx >> 5;
    if (e >= n_edges) return;
    const int d0 = (idx & 31) * 4;
    const int s = src[e];
    const int t = dst[e];

    const float4 v = *(const float4*)(T + (size_t)s * DFEAT + d0);
    unsigned* kp = keys + (size_t)t * DFEAT + d0;
    atomicMin(kp + 0, f32_to_key(v.x));
    atomicMin(kp + 1, f32_to_key(v.y));
    atomicMin(kp + 2, f32_to_key(v.z));
    atomicMin(kp + 3, f32_to_key(v.w));
}

__global__ void __launch_bounds__(256) edgeconv_finalize(
    const float4* __restrict__ A,
    float4* __restrict__ out,    // aliases the key buffer; read keys, write floats
    int n4)
{
    const int i = blockIdx.x * 256 + threadIdx.x;
    if (i >= n4) return;
    const float4 kf = out[i];
    const float4 a  = A[i];
    float4 r;
    r.x = a.x - key_to_f32(__float_as_uint(kf.x));
    r.y = a.y - key_to_f32(__float_as_uint(kf.y));
    r.z = a.z - key_to_f32(__float_as_uint(kf.z));
    r.w = a.w - key_to_f32(__float_as_uint(kf.w));
    out[i] = r;
}

// ---------------------------------------------------------------------------
extern "C" void kernel_launch(void* const* d_in, const int* in_sizes, int n_in,
                              void* d_out, int out_size, void* d_ws, size_t ws_size,
                              hipStream_t stream) {
    const float* feat    = (const float*)d_in[0];
    const float* theta_w = (const float*)d_in[1];
    const float* theta_b = (const float*)d_in[2];
    const float* phi_w   = (const float*)d_in[3];
    const float* phi_b   = (const float*)d_in[4];
    const int*   src     = (const int*)d_in[5];
    const int*   dst     = (const int*)d_in[6];

    const int n_edges = in_sizes[5];
    const int n_nodes = in_sizes[0] / DFEAT;
    const int total   = n_nodes * DFEAT;     // multiple of 4 (DFEAT = 128)
    const int total4  = total / 4;

    float* T = (float*)d_ws;                       // [N, 128]
    float* A = T + (size_t)n_nodes * DFEAT;        // [N, 128]
    unsigned* keys = (unsigned*)d_out;             // min-keys accumulate in d_out

    // 1) init min identity in d_out
    edgeconv_init_keys<<<(total4 + 255) / 256, 256, 0, stream>>>(
        (uint4*)keys, total4);

    // 2) node GEMMs (WMMA fp32): T and A
    edgeconv_node_gemm<<<(n_nodes + 15) / 16, 256, 0, stream>>>(
        feat, theta_w, theta_b, phi_w, phi_b, T, A, n_nodes);

    // 3) per-edge component-wise atomic min of T[src] into keys[dst]
    const long long ework = (long long)n_edges * 32;
    edgeconv_edge_min<<<(int)((ework + 255) / 256), 256, 0, stream>>>(
        T, src, dst, keys, n_edges);

    // 4) out = A - decoded min
    edgeconv_finalize<<<(total4 + 255) / 256, 256, 0, stream>>>(
        (const float4*)A, (float4*)d_out, total4);
}